// SelfAttention_65481071406925
// MI455X (gfx1250) — compile-verified
//
#include <hip/hip_runtime.h>
#include <math.h>

#define Bsz 4
#define Ssz 1024
#define Esz 192
#define Hsz 64
#define Dsz 3

typedef float v2f __attribute__((ext_vector_type(2)));
typedef float v4f __attribute__((ext_vector_type(4)));
typedef float v8f __attribute__((ext_vector_type(8)));

// ---------------------------------------------------------------------------
// y = x @ W^T (torch Linear). x:(M,192) W:(192,192) y:(M,192), M=4096.
// 2x2 register blocking: one wave computes a 32x32 output tile;
// per K=4 chunk: 4 b64 fragment loads feed 4 V_WMMA_F32_16X16X4_F32.
// Fragment addressing (fp32 16x4 A / 4x16 B): lane L -> row/col L%16,
// contiguous K-pair at K=(L/16)*2.
// ---------------------------------------------------------------------------
__global__ __launch_bounds__(32) void gemm_xwt_f32(const float* __restrict__ x,
                                                   const float* __restrict__ w,
                                                   float* __restrict__ y) {
  const int mt = blockIdx.x;          // 0..127  (M/32)
  const int nt = blockIdx.y;          // 0..5    (192/32)
  const int lane = threadIdx.x;
  const int rl = lane & 15;
  const int ko = (lane >> 4) << 1;    // 0 or 2

  const float* xr0 = x + (size_t)(mt * 32 + rl) * Esz;
  const float* xr1 = xr0 + (size_t)16 * Esz;
  const float* wr0 = w + (size_t)(nt * 32 + rl) * Esz;
  const float* wr1 = wr0 + (size_t)16 * Esz;

  v8f c00 = {0.f, 0.f, 0.f, 0.f, 0.f, 0.f, 0.f, 0.f};
  v8f c01 = c00, c10 = c00, c11 = c00;

#pragma unroll
  for (int kb = 0; kb < Esz; kb += 4) {
    v2f a0 = *(const v2f*)(xr0 + kb + ko);
    v2f a1 = *(const v2f*)(xr1 + kb + ko);
    v2f b0 = *(const v2f*)(wr0 + kb + ko);
    v2f b1 = *(const v2f*)(wr1 + kb + ko);
    c00 = __builtin_amdgcn_wmma_f32_16x16x4_f32(false, a0, false, b0, (short)0, c00, false, false);
    c01 = __builtin_amdgcn_wmma_f32_16x16x4_f32(false, a0, false, b1, (short)0, c01, false, false);
    c10 = __builtin_amdgcn_wmma_f32_16x16x4_f32(false, a1, false, b0, (short)0, c10, false, false);
    c11 = __builtin_amdgcn_wmma_f32_16x16x4_f32(false, a1, false, b1, (short)0, c11, false, false);
  }

  // C layout: VGPR r -> row r (lanes 0-15) / row r+8 (lanes 16-31), col = lane%16
  const int half = lane >> 4;
  const int rbase = mt * 32 + half * 8;
  const int cbase = nt * 32 + rl;
  float* y00 = y + (size_t)rbase * Esz + cbase;
#pragma unroll
  for (int r = 0; r < 8; ++r) {
    y00[(size_t)r * Esz] = c00[r];                       // rows +0..,  cols +0
    y00[(size_t)r * Esz + 16] = c01[r];                  // cols +16
    y00[(size_t)(r + 16) * Esz] = c10[r];                // rows +16
    y00[(size_t)(r + 16) * Esz + 16] = c11[r];
  }
}

// ---------------------------------------------------------------------------
// Per-head 3x3 rotation on q,k; relayout q,k,v from (B,S,E) rows to
// (B,H,S,4), 4th slot zero (K=4 zero-pad for WMMA + 16B aligned v rows).
// rot[h][d][e]: q_rot[e] = sum_d q[d]*rot[h][d][e]
// ---------------------------------------------------------------------------
__global__ __launch_bounds__(256) void rot_relayout(
    const float* __restrict__ q_raw, const float* __restrict__ k_raw,
    const float* __restrict__ v_raw, const float* __restrict__ rot,
    float* __restrict__ qr, float* __restrict__ kr, float* __restrict__ vp) {
  const int idx = blockIdx.x * blockDim.x + threadIdx.x;
  if (idx >= Bsz * Ssz * Hsz) return;
  const int h = idx & (Hsz - 1);
  const int bs = idx >> 6;
  const int b = bs >> 10;
  const int s = bs & (Ssz - 1);

  const float* qp = q_raw + (size_t)bs * Esz + h * 3;
  const float* kp = k_raw + (size_t)bs * Esz + h * 3;
  const float* vv = v_raw + (size_t)bs * Esz + h * 3;
  const float* R = rot + h * 9;

  const float q0 = qp[0], q1 = qp[1], q2 = qp[2];
  const float k0 = kp[0], k1 = kp[1], k2 = kp[2];

  const size_t o = ((size_t)(b * Hsz + h) * Ssz + s) * 4;
#pragma unroll
  for (int e = 0; e < 3; ++e) {
    qr[o + e] = q0 * R[e] + q1 * R[3 + e] + q2 * R[6 + e];
    kr[o + e] = k0 * R[e] + k1 * R[3 + e] + k2 * R[6 + e];
    vp[o + e] = vv[e];
  }
  qr[o + 3] = 0.f;
  kr[o + 3] = 0.f;
  vp[o + 3] = 0.f;
}

// ---------------------------------------------------------------------------
// Flash attention: one wave handles TWO 16-query tiles of one (b,h) so each
// loaded k-fragment / v-row feeds 2 WMMAs.  Scores via V_WMMA_F32_16X16X4_F32
// (D=3 zero-padded to K=4).  Softmax is per-lane online (each lane owns key
// columns n == lane%16 of every tile) in base-2 domain -> ZERO in-loop
// shuffles; a single cross-16-lane merge (max + rescale + sum) at the end.
// ---------------------------------------------------------------------------
__global__ __launch_bounds__(32) void attn_kernel(const float* __restrict__ qr,
                                                  const float* __restrict__ kr,
                                                  const float* __restrict__ vp,
                                                  float* __restrict__ out) {
  const int lane = threadIdx.x;
  const int bid = blockIdx.x;             // B*H*(S/32) = 8192
  const int qp2 = bid & 31;               // query tile-pair
  const int bh = bid >> 5;                // 0..255
  const int b = bh >> 6;
  const int h = bh & (Hsz - 1);

  const float* qb = qr + (size_t)bh * Ssz * 4;
  const float* kb = kr + (size_t)bh * Ssz * 4;
  const float* vb = vp + (size_t)bh * Ssz * 4;

  const int rl = lane & 15;
  const int ko = (lane >> 4) << 1;
  // (1/sqrt(3)) * log2(e): softmax done in base-2 (v_exp_f32 is native exp2)
  const float scale2 = 0.8329404f;

  const int qt0 = qp2 * 2, qt1 = qp2 * 2 + 1;
  const v2f aA = *(const v2f*)(qb + (size_t)(qt0 * 16 + rl) * 4 + ko);
  const v2f aB = *(const v2f*)(qb + (size_t)(qt1 * 16 + rl) * 4 + ko);

  float mA[8], lA[8], xA[8], yA[8], zA[8];
  float mB[8], lB[8], xB[8], yB[8], zB[8];
#pragma unroll
  for (int r = 0; r < 8; ++r) {
    mA[r] = -INFINITY; lA[r] = 0.f; xA[r] = 0.f; yA[r] = 0.f; zA[r] = 0.f;
    mB[r] = -INFINITY; lB[r] = 0.f; xB[r] = 0.f; yB[r] = 0.f; zB[r] = 0.f;
  }

  const v8f cz = {0.f, 0.f, 0.f, 0.f, 0.f, 0.f, 0.f, 0.f};

#pragma unroll 2
  for (int j = 0; j < Ssz / 16; ++j) {
    const size_t rowoff = (size_t)(j * 16 + rl) * 4;
    const v2f bf = *(const v2f*)(kb + rowoff + ko);
    const v4f vrow = *(const v4f*)(vb + rowoff);

    // prefetch a few tiles ahead (global_prefetch_b8)
    const int jp = (j + 4 < Ssz / 16) ? (j + 4) : (Ssz / 16 - 1);
    __builtin_prefetch(kb + (size_t)(jp * 16 + rl) * 4, 0, 1);
    __builtin_prefetch(vb + (size_t)(jp * 16 + rl) * 4, 0, 1);

    const v8f cA = __builtin_amdgcn_wmma_f32_16x16x4_f32(false, aA, false, bf,
                                                         (short)0, cz, false, false);
    const v8f cB = __builtin_amdgcn_wmma_f32_16x16x4_f32(false, aB, false, bf,
                                                         (short)0, cz, false, false);
#pragma unroll
    for (int r = 0; r < 8; ++r) {
      {
        const float s = cA[r] * scale2;
        const float mn = fmaxf(mA[r], s);
        const float al = exp2f(mA[r] - mn);
        const float p = exp2f(s - mn);
        mA[r] = mn;
        lA[r] = lA[r] * al + p;
        xA[r] = xA[r] * al + p * vrow.x;
        yA[r] = yA[r] * al + p * vrow.y;
        zA[r] = zA[r] * al + p * vrow.z;
      }
      {
        const float s = cB[r] * scale2;
        const float mn = fmaxf(mB[r], s);
        const float al = exp2f(mB[r] - mn);
        const float p = exp2f(s - mn);
        mB[r] = mn;
        lB[r] = lB[r] * al + p;
        xB[r] = xB[r] * al + p * vrow.x;
        yB[r] = yB[r] * al + p * vrow.y;
        zB[r] = zB[r] * al + p * vrow.z;
      }
    }
  }

  // Merge the 16 per-lane softmax partials of each row, then write.
  const int half = lane >> 4;
#pragma unroll
  for (int r = 0; r < 8; ++r) {
#pragma unroll
    for (int t = 0; t < 2; ++t) {
      float m0 = t ? mB[r] : mA[r];
      float l0 = t ? lB[r] : lA[r];
      float x0 = t ? xB[r] : xA[r];
      float y0 = t ? yB[r] : yA[r];
      float z0 = t ? zB[r] : zA[r];

      float mt = m0;
      mt = fmaxf(mt, __shfl_xor(mt, 1, 32));
      mt = fmaxf(mt, __shfl_xor(mt, 2, 32));
      mt = fmaxf(mt, __shfl_xor(mt, 4, 32));
      mt = fmaxf(mt, __shfl_xor(mt, 8, 32));
      const float sc = exp2f(m0 - mt);
      l0 *= sc; x0 *= sc; y0 *= sc; z0 *= sc;
      l0 += __shfl_xor(l0, 1, 32); l0 += __shfl_xor(l0, 2, 32);
      l0 += __shfl_xor(l0, 4, 32); l0 += __shfl_xor(l0, 8, 32);
      x0 += __shfl_xor(x0, 1, 32); x0 += __shfl_xor(x0, 2, 32);
      x0 += __shfl_xor(x0, 4, 32); x0 += __shfl_xor(x0, 8, 32);
      y0 += __shfl_xor(y0, 1, 32); y0 += __shfl_xor(y0, 2, 32);
      y0 += __shfl_xor(y0, 4, 32); y0 += __shfl_xor(y0, 8, 32);
      z0 += __shfl_xor(z0, 1, 32); z0 += __shfl_xor(z0, 2, 32);
      z0 += __shfl_xor(z0, 4, 32); z0 += __shfl_xor(z0, 8, 32);

      if (rl == r) {
        const int qt = t ? qt1 : qt0;
        const int s = qt * 16 + half * 8 + r;
        const float inv = 1.0f / l0;
        float* op = out + ((size_t)(b * Ssz + s)) * Esz + h * 3;
        op[0] = x0 * inv;
        op[1] = y0 * inv;
        op[2] = z0 * inv;
      }
    }
  }
}

// ---------------------------------------------------------------------------
extern "C" void kernel_launch(void* const* d_in, const int* in_sizes, int n_in,
                              void* d_out, int out_size, void* d_ws,
                              size_t ws_size, hipStream_t stream) {
  (void)in_sizes; (void)n_in; (void)out_size; (void)ws_size;
  const float* x   = (const float*)d_in[0];
  const float* Wq  = (const float*)d_in[1];
  const float* Wk  = (const float*)d_in[2];
  const float* Wv  = (const float*)d_in[3];
  const float* Wo  = (const float*)d_in[4];
  const float* rot = (const float*)d_in[5];
  float* out = (float*)d_out;

  const size_t NE = (size_t)Bsz * Ssz * Esz;      // 786432
  const size_t NP = (size_t)Bsz * Hsz * Ssz * 4;  // 1048576

  float* ws    = (float*)d_ws;
  float* q_raw = ws;
  float* k_raw = q_raw + NE;
  float* v_raw = k_raw + NE;
  float* qrp   = v_raw + NE;
  float* krp   = qrp + NP;
  float* vpp   = krp + NP;
  float* ao    = vpp + NP;                        // attention out (B,S,E)

  const dim3 gGemm(Bsz * Ssz / 32, Esz / 32);     // (128, 6)
  const dim3 bWave(32);

  gemm_xwt_f32<<<gGemm, bWave, 0, stream>>>(x, Wq, q_raw);
  gemm_xwt_f32<<<gGemm, bWave, 0, stream>>>(x, Wk, k_raw);
  gemm_xwt_f32<<<gGemm, bWave, 0, stream>>>(x, Wv, v_raw);

  rot_relayout<<<(Bsz * Ssz * Hsz) / 256, 256, 0, stream>>>(
      q_raw, k_raw, v_raw, rot, qrp, krp, vpp);

  attn_kernel<<<Bsz * Hsz * (Ssz / 32), bWave, 0, stream>>>(qrp, krp, vpp, ao);

  gemm_xwt_f32<<<gGemm, bWave, 0, stream>>>(ao, Wo, out);
}